// MixEHR_Seed_274877907574
// MI455X (gfx1250) — compile-verified
//
#include <hip/hip_runtime.h>
#include <math.h>

typedef float v2f __attribute__((ext_vector_type(2)));
typedef float v8f __attribute__((ext_vector_type(8)));

#define BN 64
#define VN 10000
#define KN 64
#define ETA  0.1f
#define BETA 0.05f
#define MU   0.05f
#define MINI 1e-6f

// ---- workspace layout (float offsets), total ~12.8 MB ----
#define WS_INVDS 0
#define WS_INVDN 64
#define WS_ASS   128
#define WS_ASR   (128 + VN*KN)
#define WS_ARR   (128 + 2*VN*KN)
#define WS_SEEDF (128 + 3*VN*KN)
#define WS_US    (WS_SEEDF + 10048)
#define WS_UR    (WS_US + BN*VN)

// ---- output layout (floats, concatenated tuple) ----
#define OUT_M   0
#define OUT_N   (BN*KN)
#define OUT_S   (BN*KN + VN*KN)
#define OUT_GSR (BN*KN + 2*VN*KN)
#define OUT_Q   (OUT_GSR + KN)

// Zero the atomically-accumulated output regions.
__global__ void k0_zero(float* out) {
    int i = blockIdx.x * blockDim.x + threadIdx.x;
    if (i < BN*KN) out[OUT_M + i] = 0.f;
    if (i < KN + 1) out[OUT_GSR + i] = 0.f;
}

// Column reductions over V: denom_s[k] = MU*S[k]+sum(exp_s), denom_n[k] = BETA*V+sum(exp_n).
__global__ void k1_sums(const float* __restrict__ seeds,
                        const float* __restrict__ exp_s,
                        const float* __restrict__ exp_n,
                        float* __restrict__ ws) {
    __shared__ float red[2048];
    int tid  = threadIdx.x;      // 1024 threads
    int k    = tid & 63;
    int part = tid >> 6;         // 16 partials per column
    float accs = 0.f, accn = 0.f;
    for (int v = part; v < VN; v += 16) {
        int idx = v*KN + k;
        accs += MU * seeds[idx] + exp_s[idx];
        accn += exp_n[idx];
    }
    red[tid] = accs; red[1024 + tid] = accn;
    __syncthreads();
    for (int s = 8; s >= 1; s >>= 1) {
        if (part < s) {
            red[part*64 + k]        += red[(part+s)*64 + k];
            red[1024 + part*64 + k] += red[1024 + (part+s)*64 + k];
        }
        __syncthreads();
    }
    if (part == 0) {
        ws[WS_INVDS + k] = 1.f / red[k];
        ws[WS_INVDN + k] = 1.f / (BETA * (float)VN + red[1024 + k]);
    }
}

// Per-(v,k) factors: a_ss = seed*phi_s*pi, a_sr = seed*phi_n*(1-pi), a_rr = (1-seed)*phi_n.
__global__ void k2_prep(const float* __restrict__ seeds,
                        const float* __restrict__ exp_s,
                        const float* __restrict__ exp_n,
                        const float* __restrict__ pi,
                        float* __restrict__ ws, int nwaves) {
    int lane = threadIdx.x & 31;
    int wv   = __builtin_amdgcn_readfirstlane((int)(threadIdx.x >> 5));
    int wgid = blockIdx.x * (blockDim.x >> 5) + wv;
    int k0 = lane, k1 = lane + 32;
    float ids0 = ws[WS_INVDS+k0], ids1 = ws[WS_INVDS+k1];
    float idn0 = ws[WS_INVDN+k0], idn1 = ws[WS_INVDN+k1];
    float pi0 = pi[k0], pi1 = pi[k1];
    float pm0 = 1.f - pi0, pm1 = 1.f - pi1;
    for (int v = wgid; v < VN; v += nwaves) {
        int i0 = v*KN + k0, i1 = v*KN + k1;
        float s0 = seeds[i0], s1 = seeds[i1];
        float phs0 = (MU + exp_s[i0]) * ids0;
        float phs1 = (MU + exp_s[i1]) * ids1;
        float phn0 = (BETA + exp_n[i0]) * idn0;
        float phn1 = (BETA + exp_n[i1]) * idn1;
        ws[WS_ASS+i0] = s0 * phs0 * pi0;   ws[WS_ASS+i1] = s1 * phs1 * pi1;
        ws[WS_ASR+i0] = s0 * phn0 * pm0;   ws[WS_ASR+i1] = s1 * phn1 * pm1;
        ws[WS_ARR+i0] = (1.f - s0) * phn0; ws[WS_ARR+i1] = (1.f - s1) * phn1;
        int any = __any((s0 > 0.f) || (s1 > 0.f));
        if (lane == 0) ws[WS_SEEDF + v] = any ? 1.f : 0.f;
    }
}

// WMMA GEMM: s_sum[b,v] = theta[b,:]·(a_ss+a_sr)[v,:], r_sum[b,v] = theta[b,:]·a_rr[v,:].
// Store reciprocals us = 1/(s_sum+MINI), ur = 1/(r_sum+MINI).
// V_WMMA_F32_16X16X4_F32 layouts (ISA 7.12.2):
//   A 16x4: reg0 = K={0|2}, reg1 = K={1|3}, lanes 0-15 -> M, lanes 16-31 -> M (K+2)
//   B 4x16: reg0 = K={0|2} row across lanes (N), reg1 = K={1|3}
//   C/D 16x16: reg r -> M=r (lanes 0-15) / M=r+8 (lanes 16-31), N = lane&15
__global__ void k3_wmma_sums(const float* __restrict__ exp_m,
                             float* __restrict__ ws) {
    __shared__ float th[BN*KN];
    int tid = threadIdx.x;
    for (int i = tid; i < BN*KN; i += blockDim.x) th[i] = exp_m[i] + ETA;
    __syncthreads();
    int lane = tid & 31;
    int wv   = __builtin_amdgcn_readfirstlane((int)(tid >> 5));
    int job  = blockIdx.x * (blockDim.x >> 5) + wv;
    if (job >= (BN/16)*(VN/16)) return;
    int b0 = (job & 3) * 16;
    int v0 = (job >> 2) * 16;
    int m  = lane & 15;
    int hi = lane >> 4;
    int rowt = (b0 + m) * KN;
    int rowv = (v0 + m) * KN;
    const float* ASSp = ws + WS_ASS;
    const float* ASRp = ws + WS_ASR;
    const float* ARRp = ws + WS_ARR;
    v8f cs = {0.f,0.f,0.f,0.f,0.f,0.f,0.f,0.f};
    v8f cr = cs;
    for (int kk = 0; kk < KN; kk += 4) {
        int kb = kk + 2*hi;
        v2f a;  a.x = th[rowt + kb]; a.y = th[rowt + kb + 1];
        float2 x = *(const float2*)(ASSp + rowv + kb);
        float2 y = *(const float2*)(ASRp + rowv + kb);
        float2 z = *(const float2*)(ARRp + rowv + kb);
        v2f bs; bs.x = x.x + y.x; bs.y = x.y + y.y;
        v2f br; br.x = z.x;       br.y = z.y;
        cs = __builtin_amdgcn_wmma_f32_16x16x4_f32(false, a, false, bs, (short)0, cs, false, false);
        cr = __builtin_amdgcn_wmma_f32_16x16x4_f32(false, a, false, br, (short)0, cr, false, false);
    }
    float* USp = ws + WS_US;
    float* URp = ws + WS_UR;
    int vv = v0 + m;
    int bb = b0 + hi*8;
#pragma unroll
    for (int r = 0; r < 8; ++r) {
        USp[(bb + r)*VN + vv] = 1.f / (cs[r] + MINI);
        URp[(bb + r)*VN + vv] = 1.f / (cr[r] + MINI);
    }
}

// Fused finisher: wave-per-v, uniform b-loop; all outputs produced here.
__global__ void k4_main(const float* __restrict__ bow,
                        const float* __restrict__ exp_m,
                        const float* __restrict__ pi,
                        const float* __restrict__ ws,
                        float* __restrict__ out, int nwaves) {
    __shared__ float th[BN*KN];
    __shared__ float macc[BN*KN];
    int tid = threadIdx.x;
    for (int i = tid; i < BN*KN; i += blockDim.x) { th[i] = exp_m[i] + ETA; macc[i] = 0.f; }
    __syncthreads();
    int lane = tid & 31;
    int wv   = __builtin_amdgcn_readfirstlane((int)(tid >> 5));
    int wgid = blockIdx.x * (blockDim.x >> 5) + wv;
    int k0 = lane, k1 = lane + 32;
    float pi0 = pi[k0], pi1 = pi[k1];
    float pm0 = 1.f - pi0, pm1 = 1.f - pi1;
    const float* ASSp = ws + WS_ASS;
    const float* ASRp = ws + WS_ASR;
    const float* ARRp = ws + WS_ARR;
    const float* USp  = ws + WS_US;
    const float* URp  = ws + WS_UR;
    float gs0 = 0.f, gs1 = 0.f, q = 0.f;
    for (int v = wgid; v < VN; v += nwaves) {
        int i0 = v*KN + k0, i1 = v*KN + k1;
        float ass0 = ASSp[i0], ass1 = ASSp[i1];
        float asr0 = ASRp[i0], asr1 = ASRp[i1];
        float arr0 = ARRp[i0], arr1 = ARRp[i1];
        bool seedrow = ws[WS_SEEDF + v] > 0.5f;
        float cs0 = seedrow ? (pi0*ass0 + pm0*asr0) : 0.f;
        float cs1 = seedrow ? (pi1*ass1 + pm1*asr1) : 0.f;
        float cr0 = seedrow ? pm0*arr0 : arr0;
        float cr1 = seedrow ? pm1*arr1 : arr1;
        float en0 = 0.f, en1 = 0.f, es0 = 0.f, es1 = 0.f;
        for (int b = 0; b < BN; ++b) {
            float cnt = bow[b*VN + v];               // wave-uniform -> s_load
            if (cnt == 0.f) continue;                // zero contribution everywhere
            float us = USp[b*VN + v], ur = URp[b*VN + v];
            float wts = cnt*us, wtr = cnt*ur;
            float t0 = th[b*KN + k0], t1 = th[b*KN + k1];
            float g0 = t0 * (us*cs0 + ur*cr0);       // gamma (normalized)
            float g1 = t1 * (us*cs1 + ur*cr1);
            q += g0*__logf(g0 + MINI) + g1*__logf(g1 + MINI);
            en0 += t0*(wts*asr0 + wtr*arr0);         // (g_sr+g_rr)*cnt
            en1 += t1*(wts*asr1 + wtr*arr1);
            es0 += t0*(wts*ass0);                    // g_ss*cnt
            es1 += t1*(wts*ass1);
            gs0 += t0*(us*asr0);                     // g_sr (no cnt)
            gs1 += t1*(us*asr1);
            atomicAdd(&macc[b*KN + k0], g0*cnt);
            atomicAdd(&macc[b*KN + k1], g1*cnt);
        }
        out[OUT_N + i0] = en0; out[OUT_N + i1] = en1;
        out[OUT_S + i0] = es0; out[OUT_S + i1] = es1;
    }
    atomicAdd(&out[OUT_GSR + k0], gs0);
    atomicAdd(&out[OUT_GSR + k1], gs1);
    for (int off = 16; off > 0; off >>= 1) q += __shfl_down(q, off, 32);
    if (lane == 0) atomicAdd(&out[OUT_Q], q);
    __syncthreads();
    for (int i = tid; i < BN*KN; i += blockDim.x) atomicAdd(&out[OUT_M + i], macc[i]);
}

extern "C" void kernel_launch(void* const* d_in, const int* in_sizes, int n_in,
                              void* d_out, int out_size, void* d_ws, size_t ws_size,
                              hipStream_t stream) {
    const float* bow   = (const float*)d_in[0];
    const float* seeds = (const float*)d_in[1];
    const float* exp_m = (const float*)d_in[2];
    const float* exp_s = (const float*)d_in[3];
    const float* exp_n = (const float*)d_in[4];
    const float* pi    = (const float*)d_in[5];
    float* out = (float*)d_out;
    float* ws  = (float*)d_ws;

    k0_zero<<<17, 256, 0, stream>>>(out);
    k1_sums<<<1, 1024, 0, stream>>>(seeds, exp_s, exp_n, ws);
    {
        int blocks = 125;
        k2_prep<<<blocks, 256, 0, stream>>>(seeds, exp_s, exp_n, pi, ws, blocks * 8);
    }
    {
        int jobs = (BN/16) * (VN/16);     // 2500 16x16 tiles
        int blocks = (jobs + 7) / 8;      // 8 waves per block
        k3_wmma_sums<<<blocks, 256, 0, stream>>>(exp_m, ws);
    }
    {
        int blocks = 250;
        k4_main<<<blocks, 256, 0, stream>>>(bow, exp_m, pi, ws, out, blocks * 8);
    }
}